// GraphSAGE_18640158065248
// MI455X (gfx1250) — compile-verified
//
#include <hip/hip_runtime.h>
#include <math.h>

typedef float v2f __attribute__((ext_vector_type(2)));
typedef float v8f __attribute__((ext_vector_type(8)));

#define NF 64   // NFEAT == NHID == 64

// ---------------------------------------------------------------- utilities
__global__ void zero_f32_kernel(float* __restrict__ p, long long n) {
  long long i = (long long)blockIdx.x * blockDim.x + threadIdx.x;
  if (i < n) p[i] = 0.0f;
}

__global__ void degree_kernel(const long long* __restrict__ dst,
                              float* __restrict__ deg, int nE) {
  int e = blockIdx.x * blockDim.x + threadIdx.x;
  if (e < nE) atomicAdd(&deg[dst[e]], 1.0f);
}

__global__ void invdeg_kernel(float* __restrict__ deg, int n) {
  int i = blockIdx.x * blockDim.x + threadIdx.x;
  if (i < n) deg[i] = 1.0f / fmaxf(deg[i], 1.0f);
}

// One thread per (edge, feature): agg[dst] += feat[src]. L2-resident scatter.
__global__ void scatter_feat_kernel(const float* __restrict__ feat,
                                    const long long* __restrict__ src,
                                    const long long* __restrict__ dst,
                                    float* __restrict__ agg,
                                    long long nEdgeFeat) {
  long long tid = (long long)blockIdx.x * blockDim.x + threadIdx.x;
  if (tid >= nEdgeFeat) return;
  long long e = tid >> 6;          // 64 features per edge
  int f = (int)(tid & 63);
  long long s = src[e];
  long long d = dst[e];
  atomicAdd(&agg[d * NF + f], feat[s * NF + f]);
}

// ------------------------------------------------- layer 1: fused dual GEMM
// h = (agg*invdeg) @ Wl^T + bl + x @ Wr^T   (all fp32, V_WMMA_F32_16X16X4_F32)
// One wave = one 16-row node tile, 4 accumulators cover all 64 output cols.
__global__ void __launch_bounds__(128)
sage_layer1_kernel(const float* __restrict__ agg, const float* __restrict__ dinv,
                   const float* __restrict__ x,
                   const float* __restrict__ Wl, const float* __restrict__ bl,
                   const float* __restrict__ Wr,
                   float* __restrict__ hout, int nNodes) {
  const int lane = threadIdx.x & 31;
  const int wave = blockIdx.x * (blockDim.x >> 5) + (threadIdx.x >> 5);
  const int m0 = wave * 16;
  if (m0 >= nNodes) return;                 // whole-wave exit: EXEC stays all-1s

  const int mrow  = m0 + (lane & 15);       // A row for this lane
  const int khalf = (lane >> 4) << 1;       // K offset 0 or 2 per lane half
  const int ncol  = lane & 15;              // B/C/D column for this lane
  const float inv = dinv[mrow];
  const float* aggRow = agg + (long long)mrow * NF;
  const float* xRow   = x   + (long long)mrow * NF;

  v8f acc0 = {}, acc1 = {}, acc2 = {}, acc3 = {};

#pragma unroll
  for (int k0 = 0; k0 < NF; k0 += 4) {
    const int ka = k0 + khalf;
    v2f aM, aX;
    aM[0] = aggRow[ka] * inv; aM[1] = aggRow[ka + 1] * inv;   // mean on the fly
    aX[0] = xRow[ka];         aX[1] = xRow[ka + 1];
#define WSTEP(ACC, N0) { \
    v2f bL, bR; \
    bL[0] = Wl[(N0 + ncol) * NF + ka]; bL[1] = Wl[(N0 + ncol) * NF + ka + 1]; \
    bR[0] = Wr[(N0 + ncol) * NF + ka]; bR[1] = Wr[(N0 + ncol) * NF + ka + 1]; \
    ACC = __builtin_amdgcn_wmma_f32_16x16x4_f32(false, aM, false, bL, (short)0, ACC, false, false); \
    ACC = __builtin_amdgcn_wmma_f32_16x16x4_f32(false, aX, false, bR, (short)0, ACC, false, false); }
    WSTEP(acc0, 0) WSTEP(acc1, 16) WSTEP(acc2, 32) WSTEP(acc3, 48)
#undef WSTEP
  }

  const int mBase = m0 + ((lane >> 4) << 3);  // C/D: VGPR v -> row mBase+v
#define WOUT(ACC, N0) { \
    const float b = bl[N0 + ncol]; \
    _Pragma("unroll") \
    for (int v = 0; v < 8; ++v) \
      hout[(long long)(mBase + v) * NF + N0 + ncol] = ACC[v] + b; }
  WOUT(acc0, 0) WOUT(acc1, 16) WOUT(acc2, 32) WOUT(acc3, 48)
#undef WOUT
}

// --------------------- layer 2: dual GEMM (N=2 padded to 16) + log_softmax
__global__ void __launch_bounds__(128)
sage_layer2_kernel(const float* __restrict__ agg, const float* __restrict__ dinv,
                   const float* __restrict__ h,
                   const float* __restrict__ Wl, const float* __restrict__ bl,
                   const float* __restrict__ Wr,
                   float* __restrict__ out, int nNodes) {
  const int lane = threadIdx.x & 31;
  const int wave = blockIdx.x * (blockDim.x >> 5) + (threadIdx.x >> 5);
  const int m0 = wave * 16;
  if (m0 >= nNodes) return;

  const int mrow  = m0 + (lane & 15);
  const int khalf = (lane >> 4) << 1;
  const int ncol  = lane & 15;
  const bool valid = (ncol < 2);            // only 2 real output columns
  const float inv = dinv[mrow];
  const float* aggRow = agg + (long long)mrow * NF;
  const float* hRow   = h   + (long long)mrow * NF;

  v8f acc = {};
#pragma unroll
  for (int k0 = 0; k0 < NF; k0 += 4) {
    const int ka = k0 + khalf;
    v2f aM, aX, bL = {0.0f, 0.0f}, bR = {0.0f, 0.0f};
    aM[0] = aggRow[ka] * inv; aM[1] = aggRow[ka + 1] * inv;
    aX[0] = hRow[ka];         aX[1] = hRow[ka + 1];
    if (valid) {                            // pad columns 2..15 with zeros
      bL[0] = Wl[ncol * NF + ka]; bL[1] = Wl[ncol * NF + ka + 1];
      bR[0] = Wr[ncol * NF + ka]; bR[1] = Wr[ncol * NF + ka + 1];
    }
    acc = __builtin_amdgcn_wmma_f32_16x16x4_f32(false, aM, false, bL, (short)0, acc, false, false);
    acc = __builtin_amdgcn_wmma_f32_16x16x4_f32(false, aX, false, bR, (short)0, acc, false, false);
  }

  const float b = valid ? bl[ncol] : 0.0f;
  const int mBase = m0 + ((lane >> 4) << 3);
#pragma unroll
  for (int v = 0; v < 8; ++v) {
    float l = acc[v] + b;                   // logit for (row mBase+v, col ncol)
    float o = __shfl_xor(l, 1, 32);         // partner logit (lane pairs 0-1, 16-17)
    float mx  = fmaxf(l, o);
    float lse = mx + logf(expf(l - mx) + expf(o - mx));
    if (valid) out[(long long)(mBase + v) * 2 + ncol] = l - lse;
  }
}

// ---------------------------------------------------------------- launcher
extern "C" void kernel_launch(void* const* d_in, const int* in_sizes, int n_in,
                              void* d_out, int out_size, void* d_ws, size_t ws_size,
                              hipStream_t stream) {
  (void)n_in; (void)out_size; (void)ws_size;
  const float*     x   = (const float*)d_in[0];
  const long long* ei  = (const long long*)d_in[1];   // int64 per reference
  const float*     Wl1 = (const float*)d_in[2];
  const float*     bl1 = (const float*)d_in[3];
  const float*     Wr1 = (const float*)d_in[4];
  const float*     Wl2 = (const float*)d_in[5];
  const float*     bl2 = (const float*)d_in[6];
  const float*     Wr2 = (const float*)d_in[7];

  const int nNodes = in_sizes[0] / NF;
  const int nEdges = in_sizes[1] / 2;
  const long long* src = ei;            // edge_index[0, :]
  const long long* dst = ei + nEdges;   // edge_index[1, :]

  // workspace: agg [N*64] | h [N*64] | invdeg [N]   (~52 MB)
  float* agg  = (float*)d_ws;
  float* h    = agg + (long long)nNodes * NF;
  float* dinv = h   + (long long)nNodes * NF;

  const long long nAgg = (long long)nNodes * NF;
  const long long nEF  = (long long)nEdges * NF;

  // degrees (shared by both layers) and reciprocal
  zero_f32_kernel<<<(unsigned)((nAgg + 255) / 256), 256, 0, stream>>>(agg, nAgg);
  zero_f32_kernel<<<(unsigned)((nNodes + 255) / 256), 256, 0, stream>>>(dinv, (long long)nNodes);
  degree_kernel<<<(nEdges + 255) / 256, 256, 0, stream>>>(dst, dinv, nEdges);
  invdeg_kernel<<<(nNodes + 255) / 256, 256, 0, stream>>>(dinv, nNodes);

  // layer 1
  scatter_feat_kernel<<<(unsigned)((nEF + 255) / 256), 256, 0, stream>>>(x, src, dst, agg, nEF);
  const int nTiles = (nNodes + 15) / 16;
  const int gemmBlocks = (nTiles + 3) / 4;            // 4 waves / block (wave32)
  sage_layer1_kernel<<<gemmBlocks, 128, 0, stream>>>(agg, dinv, x, Wl1, bl1, Wr1, h, nNodes);

  // layer 2
  zero_f32_kernel<<<(unsigned)((nAgg + 255) / 256), 256, 0, stream>>>(agg, nAgg);
  scatter_feat_kernel<<<(unsigned)((nEF + 255) / 256), 256, 0, stream>>>(h, src, dst, agg, nEF);
  sage_layer2_kernel<<<gemmBlocks, 128, 0, stream>>>(agg, dinv, h, Wl2, bl2, Wr2,
                                                     (float*)d_out, nNodes);
}